// DetectionLayer_43052752175797
// MI455X (gfx1250) — compile-verified
//
#include <hip/hip_runtime.h>
#include <math.h>

#define NCLS  80
#define GS    52
#define BSZ   32
#define NGT   20
#define NA    3
#define NCH   85                 // 5 + NCLS
#define CELLS (GS * GS)          // 2704
#define NREC  (BSZ * NGT)        // 640
#define NTOT  (BSZ * NA * CELLS) // 259584

typedef __attribute__((ext_vector_type(2))) float v2f;
typedef __attribute__((ext_vector_type(8))) float v8f;

__device__ __forceinline__ float sigm(float v) { return 1.0f / (1.0f + expf(-v)); }

// ---------------------------------------------------------------------------
// Kernel A: one wave, lane = batch. Serial GT loop reproduces the scatter's
// last-write-wins dedup deterministically. Emits 640 records:
//   lins[r]  = linear cell index ((b*3+a)*52+gj)*52+gi, or -1 if overwritten
//   gtv[r*6] = {gt_tx, gt_ty, gt_tw, gt_th, best_iou, class_id}
// ---------------------------------------------------------------------------
__global__ void yolo_assign(const float* __restrict__ x,
                            const float* __restrict__ yt,
                            const float* __restrict__ anc,
                            int*   __restrict__ lins,
                            float* __restrict__ gtv) {
  const int b = threadIdx.x;
  if (b >= BSZ) return;
  float sa[NA][2];
#pragma unroll
  for (int a = 0; a < NA; ++a) {
    sa[a][0] = anc[a * 2 + 0] * 0.125f;   // stride = 416/52 = 8
    sa[a][1] = anc[a * 2 + 1] * 0.125f;
  }
  int lin_loc[NGT];
  for (int n = 0; n < NGT; ++n) {
    const float* y = yt + (b * NGT + n) * 5;
    const float gx = y[0] * (float)GS, gy = y[1] * (float)GS;
    const float gw = y[2] * (float)GS, gh = y[3] * (float)GS;
    int gi = (int)gx; gi = gi < 0 ? 0 : (gi > GS - 1 ? GS - 1 : gi);
    int gj = (int)gy; gj = gj < 0 ? 0 : (gj > GS - 1 ? GS - 1 : gj);
    const float gx0 = gx - 0.5f * gw, gy0 = gy - 0.5f * gh;
    const float gx1 = gx + 0.5f * gw, gy1 = gy + 0.5f * gh;
    const float garea = (gx1 - gx0) * (gy1 - gy0);
    float best_iou = -1.0f; int best_a = 0;
#pragma unroll
    for (int a = 0; a < NA; ++a) {
      const int base = (b * (NA * NCH) + a * NCH) * CELLS + gj * GS + gi;
      const float tx = sigm(x[base + 0 * CELLS]);
      const float ty = sigm(x[base + 1 * CELLS]);
      const float bw = expf(x[base + 2 * CELLS]) * sa[a][0];
      const float bh = expf(x[base + 3 * CELLS]) * sa[a][1];
      const float bx = tx + (float)gi, by = ty + (float)gj;
      const float ax0 = bx - 0.5f * bw, ay0 = by - 0.5f * bh;
      const float ax1 = bx + 0.5f * bw, ay1 = by + 0.5f * bh;
      const float iw = fmaxf(fminf(ax1, gx1) - fmaxf(ax0, gx0), 0.0f);
      const float ih = fmaxf(fminf(ay1, gy1) - fmaxf(ay0, gy0), 0.0f);
      const float inter = iw * ih;
      const float aarea = (ax1 - ax0) * (ay1 - ay0);
      const float iou = inter / (aarea + garea - inter + 1e-16f);
      if (iou > best_iou) { best_iou = iou; best_a = a; }  // first-max on ties
    }
    lin_loc[n] = ((b * NA + best_a) * GS + gj) * GS + gi;
    float* g = gtv + (b * NGT + n) * 6;
    g[0] = gx - (float)gi;
    g[1] = gy - (float)gj;
    g[2] = logf(gw / sa[best_a][0]);
    g[3] = logf(gh / sa[best_a][1]);
    g[4] = best_iou;
    g[5] = y[4];
  }
  for (int n = 0; n < NGT; ++n) {            // later GT wins on duplicate cell
    bool dead = false;
    for (int n2 = n + 1; n2 < NGT; ++n2)
      if (lin_loc[n2] == lin_loc[n]) dead = true;
    lins[b * NGT + n] = dead ? -1 : lin_loc[n];
  }
}

// ---------------------------------------------------------------------------
// Kernel B: one wave per record. Lanes cooperate on the 80-class log-sum-exp.
// Writes 16 component rows of pcm[16][NREC] (rows 7..15 zero-padding so the
// WMMA reduction can load unconditionally).
//   row 0..3: (tx-gt_tx)^2, (ty-gt_ty)^2, (tw-gt_tw)^2, (th-gt_th)^2
//   row 4:    CE = lse - logit[label]
//   row 5:    25*(conf-best_iou)^2
//   row 6:    live flag (1.0)
// ---------------------------------------------------------------------------
__global__ void yolo_partials(const float* __restrict__ x,
                              const int*   __restrict__ lins,
                              const float* __restrict__ gtv,
                              float* __restrict__ pcm) {
  const int r = blockIdx.x;
  const int lane = threadIdx.x;
  if (lane >= 7 && lane < 16) pcm[lane * NREC + r] = 0.0f;  // pad rows
  const int lin = lins[r];
  if (lin < 0) {
    if (lane == 0) {
#pragma unroll
      for (int m = 0; m < 7; ++m) pcm[m * NREC + r] = 0.0f;
    }
    return;
  }
  const int gi = lin % GS;
  const int gj = (lin / GS) % GS;
  const int a  = (lin / CELLS) % NA;
  const int b  = lin / (CELLS * NA);
  const int base = (b * (NA * NCH) + a * NCH) * CELLS + gj * GS + gi;

  // classes: lane, lane+32, and lane+64 (valid for lane<16)  -> 80 total
  const float v0 = x[base + (5 + lane) * CELLS];
  const float v1 = x[base + (5 + lane + 32) * CELLS];
  const float v2 = (lane < 16) ? x[base + (5 + lane + 64) * CELLS] : -3.0e38f;
  float mx = fmaxf(fmaxf(v0, v1), v2);
#pragma unroll
  for (int off = 16; off >= 1; off >>= 1) mx = fmaxf(mx, __shfl_xor(mx, off, 32));
  float s = expf(v0 - mx) + expf(v1 - mx) + ((lane < 16) ? expf(v2 - mx) : 0.0f);
#pragma unroll
  for (int off = 16; off >= 1; off >>= 1) s += __shfl_xor(s, off, 32);

  if (lane == 0) {
    const float* g = gtv + r * 6;
    const float lse = mx + logf(s);
    const int label = (int)g[5];
    const float ce = lse - x[base + (5 + label) * CELLS];
    const float tx = sigm(x[base + 0 * CELLS]);
    const float ty = sigm(x[base + 1 * CELLS]);
    const float tw = x[base + 2 * CELLS];
    const float th = x[base + 3 * CELLS];
    const float cf = sigm(x[base + 4 * CELLS]);
    const float dx = tx - g[0], dy = ty - g[1];
    const float dw = tw - g[2], dh = th - g[3];
    const float dc = 5.0f * (cf - g[4]);
    pcm[0 * NREC + r] = dx * dx;
    pcm[1 * NREC + r] = dy * dy;
    pcm[2 * NREC + r] = dw * dw;
    pcm[3 * NREC + r] = dh * dh;
    pcm[4 * NREC + r] = ce;
    pcm[5 * NREC + r] = dc * dc;
    pcm[6 * NREC + r] = 1.0f;
  }
}

// ---------------------------------------------------------------------------
// Kernel C: single wave. Exact deterministic reduction of pcm[16][640] via
// V_WMMA_F32_16X16X4_F32 with B = ones: D[m][n] += sum_k A[m][k]*1.
// A layout (16x4 f32): lane<16 -> (M=lane, K={0,1}); lane>=16 -> (M=lane-16,
// K={2,3}). After 160 steps, lane 0's v8f C holds D[0..7][0] = the 8 row sums.
// ---------------------------------------------------------------------------
__global__ void yolo_reduce(const float* __restrict__ pcm,
                            float* __restrict__ out) {
  const int lane = threadIdx.x;
  const int m = lane & 15;
  const int kbase = (lane < 16) ? 0 : 2;
  v8f c = {0.f, 0.f, 0.f, 0.f, 0.f, 0.f, 0.f, 0.f};
#if defined(__gfx1250__) && __has_builtin(__builtin_amdgcn_wmma_f32_16x16x4_f32)
  v2f bones; bones.x = 1.0f; bones.y = 1.0f;
  const float* rowp = pcm + m * NREC;
#pragma unroll 4
  for (int it = 0; it < NREC / 4; ++it) {
    const int i0 = it * 4 + kbase;
    v2f av; av.x = rowp[i0]; av.y = rowp[i0 + 1];
    c = __builtin_amdgcn_wmma_f32_16x16x4_f32(
        /*neg_a=*/false, av, /*neg_b=*/false, bones,
        /*c_mod=*/(short)0, c, /*reuse_a=*/false, /*reuse_b=*/false);
  }
#else
  if (lane == 0) {
    for (int mm = 0; mm < 8; ++mm) {
      float acc = 0.0f;
      for (int i = 0; i < NREC; ++i) acc += pcm[mm * NREC + i];
      c[mm] = acc;
    }
  }
#endif
  if (lane == 0) {
    out[0] = c[0];                                     // lx
    out[1] = c[1];                                     // ly
    out[2] = c[2];                                     // lw
    out[3] = c[3];                                     // lh
    out[4] = c[4] + ((float)NTOT - c[6]) * logf(80.0f); // lcls (+ bg rows)
    out[5] = c[5];                                     // lconf
  }
}

// ---------------------------------------------------------------------------
extern "C" void kernel_launch(void* const* d_in, const int* in_sizes, int n_in,
                              void* d_out, int out_size, void* d_ws, size_t ws_size,
                              hipStream_t stream) {
  (void)in_sizes; (void)n_in; (void)out_size; (void)ws_size;
  const float* x   = (const float*)d_in[0];
  const float* yt  = (const float*)d_in[1];
  const float* anc = (const float*)d_in[2];
  float* out = (float*)d_out;

  char* ws = (char*)d_ws;
  int*   lins = (int*)ws;                                   //  640 * 4  B
  float* gtv  = (float*)(ws + NREC * sizeof(int));          //  640 * 24 B
  float* pcm  = (float*)(ws + NREC * sizeof(int) + NREC * 6 * sizeof(float)); // 16*640*4 B

  yolo_assign  <<<1,    32, 0, stream>>>(x, yt, anc, lins, gtv);
  yolo_partials<<<NREC, 32, 0, stream>>>(x, lins, gtv, pcm);
  yolo_reduce  <<<1,    32, 0, stream>>>(pcm, out);
}